// GraphConvolution_16578573762726
// MI455X (gfx1250) — compile-verified
//
#include <hip/hip_runtime.h>
#include <hip/hip_bf16.h>

// GCN layer for MI455X (gfx1250, wave32):
//   support = X(65536x128) @ W(128x64)        -> V_WMMA_F32_16X16X4_F32 (exact f32)
//   out[r]  += edge_val * support[c]  (atomics, L2-resident: 16MB accumulator)
//   out      = prior * out + bias     (elementwise, float4)
// Memory-bound: ~600MB total traffic, support+accumulator fit in 192MB L2.

typedef __attribute__((ext_vector_type(2))) float v2f;
typedef __attribute__((ext_vector_type(8))) float v8f;

#define IN_DIM 128
#define OUT_DIM 64

// ---------------------------------------------------------------------------
// Kernel 1: support = X @ W via f32 WMMA.
// Block = 256 threads = 8 waves. Each wave computes one 16x16 tile:
//   waves 0..3 -> M-tile 0, N-tiles 0..3 ; waves 4..7 -> M-tile 1.
// Block covers 32 rows x 64 cols; grid = N_NODES/32 blocks (exact, no bounds).
// ---------------------------------------------------------------------------
__global__ __launch_bounds__(256) void gcn_gemm_wmma(
    const float* __restrict__ X,   // [N, 128]
    const float* __restrict__ W,   // [128, 64]
    float* __restrict__ S)         // [N, 64] support out
{
    const int wave  = threadIdx.x >> 5;        // 0..7
    const int lane  = threadIdx.x & 31;
    const int mtile = blockIdx.x * 2 + (wave >> 2);
    const int ntile = wave & 3;

    const int lane15 = lane & 15;
    const int m      = mtile * 16 + lane15;    // A row owned by this lane
    const int n      = ntile * 16 + lane15;    // B/C column owned by this lane
    const int khalf  = (lane >> 4) << 1;       // 0 for lanes 0-15, 2 for 16-31

    const float* xrow = X + (size_t)m * IN_DIM;

    v8f c = {};
#pragma unroll
    for (int kk = 0; kk < IN_DIM; kk += 4) {
        // A 16x4 f32 fragment: lane holds X[m][kk+khalf], X[m][kk+khalf+1] (contiguous)
        v2f a = *(const v2f*)(xrow + kk + khalf);
        // B 4x16 f32 fragment: lane holds W[kk+khalf][n], W[kk+khalf+1][n]
        v2f b;
        b.x = W[(size_t)(kk + khalf)     * OUT_DIM + n];
        b.y = W[(size_t)(kk + khalf + 1) * OUT_DIM + n];
        // (neg_a, A, neg_b, B, c_mod, C, reuse_a, reuse_b)
        c = __builtin_amdgcn_wmma_f32_16x16x4_f32(
                false, a, false, b, (short)0, c, false, false);
    }

    // D layout: VGPR r -> M = mtile*16 + r + 8*(lane>=16), N = ntile*16 + (lane&15)
    const int mrow = mtile * 16 + ((lane >> 4) << 3);
#pragma unroll
    for (int r = 0; r < 8; ++r) {
        S[(size_t)(mrow + r) * OUT_DIM + n] = c[r];
    }
}

// ---------------------------------------------------------------------------
// Kernel 2: edge scatter. 16 threads per edge, 4 cols each (float4 gather,
// 4x global_atomic_add_f32 scatter). Gather of one edge row = 256B coalesced.
// ---------------------------------------------------------------------------
__global__ __launch_bounds__(256) void gcn_scatter(
    const int*   __restrict__ erow,
    const int*   __restrict__ ecol,
    const float* __restrict__ eval_,
    const float* __restrict__ S,
    float*       __restrict__ out)
{
    const unsigned tid = blockIdx.x * 256u + threadIdx.x;
    const unsigned e   = tid >> 4;          // edge index
    const unsigned c4  = (tid & 15u) << 2;  // column offset (0,4,...,60)

    const int   r  = erow[e];
    const int   cl = ecol[e];
    const float v  = eval_[e];

    const float4 s = *(const float4*)(S + (size_t)cl * OUT_DIM + c4);
    float* o = out + (size_t)r * OUT_DIM + c4;
    atomicAdd(o + 0, v * s.x);
    atomicAdd(o + 1, v * s.y);
    atomicAdd(o + 2, v * s.z);
    atomicAdd(o + 3, v * s.w);
}

// ---------------------------------------------------------------------------
// Kernel 3: out = prior * out + bias, in place, float4 per thread.
// ---------------------------------------------------------------------------
__global__ __launch_bounds__(256) void gcn_finalize(
    float*       __restrict__ out,
    const float* __restrict__ prior,
    const float* __restrict__ bias)
{
    const size_t i  = (size_t)blockIdx.x * 256u + threadIdx.x;
    const float4 o  = *(const float4*)(out   + i * 4);
    const float4 p  = *(const float4*)(prior + i * 4);
    const float4 b  = *(const float4*)(bias + (((unsigned)i & 15u) << 2));
    float4 rr;
    rr.x = fmaf(p.x, o.x, b.x);
    rr.y = fmaf(p.y, o.y, b.y);
    rr.z = fmaf(p.z, o.z, b.z);
    rr.w = fmaf(p.w, o.w, b.w);
    *(float4*)(out + i * 4) = rr;
}

extern "C" void kernel_launch(void* const* d_in, const int* in_sizes, int n_in,
                              void* d_out, int out_size, void* d_ws, size_t ws_size,
                              hipStream_t stream) {
    const float* X     = (const float*)d_in[0];  // [N, 128]
    const int*   erow  = (const int*)  d_in[1];  // [E]
    const int*   ecol  = (const int*)  d_in[2];  // [E]
    const float* eval_ = (const float*)d_in[3];  // [E]
    const float* prior = (const float*)d_in[4];  // [N, 64]
    const float* W     = (const float*)d_in[5];  // [128, 64]
    const float* bias  = (const float*)d_in[6];  // [64]

    float* out     = (float*)d_out;              // [N, 64], doubles as accumulator
    float* support = (float*)d_ws;               // [N, 64] scratch (16 MB)

    const int N = in_sizes[0] / IN_DIM;          // 65536
    const int E = in_sizes[1];                   // 1048576

    // Zero the accumulator (d_out) — graph-capture-safe async memset.
    hipMemsetAsync(out, 0, (size_t)out_size * sizeof(float), stream);

    // 1) support = X @ W  (WMMA f32, 32 rows per block)
    gcn_gemm_wmma<<<N / 32, 256, 0, stream>>>(X, W, support);

    // 2) atomic scatter-add over edges (16 threads/edge -> E/16 blocks of 256)
    gcn_scatter<<<E / 16, 256, 0, stream>>>(erow, ecol, eval_, support, out);

    // 3) out = prior * out + bias  (N*64/4 threads)
    gcn_finalize<<<(N * OUT_DIM) / (256 * 4), 256, 0, stream>>>(out, prior, bias);
}